// MemoryAggregator_14207751815904
// MI455X (gfx1250) — compile-verified
//
#include <hip/hip_runtime.h>
#include <hip/hip_bf16.h>
#include <math.h>

// ---------------------------------------------------------------------------
// Types for CDNA5 WMMA (wave32, gfx1250)
// ---------------------------------------------------------------------------
typedef __attribute__((ext_vector_type(2))) float v2f;
typedef __attribute__((ext_vector_type(8))) float v8f;

#define INPUT_SIZE 256
#define HEAD_SIZE  32
#define INV_DK     0.17677669529663687f   // 1/sqrt(32)
#define MTILES     4                      // 16-row sub-tiles per wave (M-blocking)

// Order-preserving encoding of float into uint32 so unsigned atomicMax gives
// float max (handles negatives and -inf correctly).
__device__ __forceinline__ unsigned enc_f32(float f) {
    unsigned u = __float_as_uint(f);
    return (u & 0x80000000u) ? ~u : (u | 0x80000000u);
}
__device__ __forceinline__ float dec_f32(unsigned e) {
    unsigned u = (e & 0x80000000u) ? (e ^ 0x80000000u) : ~e;
    return __uint_as_float(u);
}
#define ENC_NEG_INF 0x007FFFFFu   // enc_f32(-inf)

// ---------------------------------------------------------------------------
// Kernel 0: initialize output, denom, rowmax
// ---------------------------------------------------------------------------
__global__ void gat_init(float* __restrict__ out, float* __restrict__ denom,
                         unsigned* __restrict__ rmax, int outN, int N) {
    int i = blockIdx.x * 256 + threadIdx.x;
    if (i < outN) out[i] = 0.0f;
    if (i < N) { denom[i] = 0.0f; rmax[i] = ENC_NEG_INF; }
}

// ---------------------------------------------------------------------------
// Kernel 1: fused QKV projection with exact-fp32 WMMA (V_WMMA_F32_16X16X4_F32)
// Each wave computes 4 consecutive 16-row tiles of X (64 node rows) against
// all 96 output columns (Q|K|V as six 16x16 f32 tiles each). The six B
// fragments per K-step are reused across the 4 row sub-tiles, so every K-step
// issues 4 b64 A-loads + 12 scalar B-loads for 24 WMMAs.
//
// fp32 A 16x4 layout : VGPR0 = K {0|2} by lane half, VGPR1 = K {1|3}; M = lane&15
// fp32 B 4x16 layout : mirrored, N = lane&15
// fp32 C/D 16x16     : VGPR j -> M = j + 8*(lane>=16), N = lane&15
// ---------------------------------------------------------------------------
__global__ __launch_bounds__(256) void gat_qkv_wmma(
    const float* __restrict__ X,
    const float* __restrict__ Wq, const float* __restrict__ Wk,
    const float* __restrict__ Wv,
    float* __restrict__ Q, float* __restrict__ Ko, float* __restrict__ Vo,
    int N)
{
    const int lane    = threadIdx.x & 31;
    const int wave    = threadIdx.x >> 5;
    const int tile0   = (blockIdx.x * 8 + wave) * MTILES;  // first 16-row tile
    if (tile0 * 16 >= N) return;            // wave-uniform: EXEC stays all-1s

    const int m  = lane & 15;               // A-matrix row / B,C,D column
    const int kh = (lane >> 4) * 2;         // K sub-offset: 0 or 2 by lane half

    const float* Ws[3] = { Wq, Wk, Wv };

    v8f acc[MTILES][6];                     // [subtile][mat*2 + colHalf]
    #pragma unroll
    for (int t = 0; t < MTILES; ++t)
        #pragma unroll
        for (int i = 0; i < 6; ++i) {
            v8f z = {};
            acc[t][i] = z;
        }

    // Per-sub-tile X row pointers; invalid sub-tiles clamp to row 0 (loads
    // stay in-bounds, results discarded at store time). valid[] is
    // lane-independent -> wave-uniform control flow only.
    const float* xrow[MTILES];
    bool valid[MTILES];
    #pragma unroll
    for (int t = 0; t < MTILES; ++t) {
        const int rb = (tile0 + t) * 16;
        valid[t] = (rb + 16) <= N;
        const int r = valid[t] ? (rb + m) : m;
        xrow[t] = X + (size_t)r * INPUT_SIZE;
    }

    #pragma unroll 4
    for (int k0 = 0; k0 < INPUT_SIZE; k0 += 4) {
        v2f a[MTILES];
        #pragma unroll
        for (int t = 0; t < MTILES; ++t) {
            a[t].x = xrow[t][k0 + kh];
            a[t].y = xrow[t][k0 + kh + 1];
        }
        #pragma unroll
        for (int w = 0; w < 3; ++w) {
            const float* W = Ws[w];
            #pragma unroll
            for (int h = 0; h < 2; ++h) {
                const int col = m + h * 16;
                v2f b;
                b.x = W[(k0 + kh)     * HEAD_SIZE + col];
                b.y = W[(k0 + kh + 1) * HEAD_SIZE + col];
                #pragma unroll
                for (int t = 0; t < MTILES; ++t) {
                    acc[t][w * 2 + h] = __builtin_amdgcn_wmma_f32_16x16x4_f32(
                        /*neg_a=*/false, a[t], /*neg_b=*/false, b,
                        /*c_mod=*/(short)0, acc[t][w * 2 + h],
                        /*reuse_a=*/false, /*reuse_b=*/false);
                }
            }
        }
    }

    float* Outs[3] = { Q, Ko, Vo };
    const int rofs = (lane >> 4) * 8;       // C/D: upper lanes hold rows M+8
    #pragma unroll
    for (int t = 0; t < MTILES; ++t) {
        if (!valid[t]) continue;            // wave-uniform branch
        const int row_base = (tile0 + t) * 16;
        #pragma unroll
        for (int w = 0; w < 3; ++w) {
            #pragma unroll
            for (int h = 0; h < 2; ++h) {
                const v8f c = acc[t][w * 2 + h];
                const int col = m + h * 16;
                #pragma unroll
                for (int j = 0; j < 8; ++j) {
                    const int row = row_base + j + rofs;
                    Outs[w][(size_t)row * HEAD_SIZE + col] = c[j];
                }
            }
        }
    }
}

// ---------------------------------------------------------------------------
// Kernel 2: per-edge scores = dot(Q[src], K[dst]) / sqrt(32); segment max via
// encoded unsigned atomicMax. 8 lanes per edge, float4 (b128) gathers.
// ---------------------------------------------------------------------------
__global__ __launch_bounds__(256) void gat_scores(
    const float* __restrict__ Q, const float* __restrict__ K,
    const int* __restrict__ src, const int* __restrict__ dst,
    float* __restrict__ scores, unsigned* __restrict__ rmax, int E)
{
    const int e = blockIdx.x * 32 + (threadIdx.x >> 3);
    if (e >= E) return;
    const int sub = threadIdx.x & 7;
    const int s = src[e];
    const int d = dst[e];

    const float4 q = ((const float4*)(Q + (size_t)s * HEAD_SIZE))[sub];
    const float4 k = ((const float4*)(K + (size_t)d * HEAD_SIZE))[sub];
    float p = q.x * k.x + q.y * k.y + q.z * k.z + q.w * k.w;
    p += __shfl_xor(p, 1, 8);
    p += __shfl_xor(p, 2, 8);
    p += __shfl_xor(p, 4, 8);

    if (sub == 0) {
        const float sc = p * INV_DK;
        scores[e] = sc;
        atomicMax(&rmax[s], enc_f32(sc));
    }
}

// ---------------------------------------------------------------------------
// Kernel 3: ex = exp(score - rowmax[src]); denom[src] += ex (in place).
// ---------------------------------------------------------------------------
__global__ __launch_bounds__(256) void gat_exp(
    const int* __restrict__ src, float* __restrict__ scores,
    const unsigned* __restrict__ rmax, float* __restrict__ denom, int E)
{
    const int e = blockIdx.x * 256 + threadIdx.x;
    if (e >= E) return;
    const int s = src[e];
    const float ex = __expf(scores[e] - dec_f32(rmax[s]));
    scores[e] = ex;                 // reuse buffer as ex[]
    atomicAdd(&denom[s], ex);
}

// ---------------------------------------------------------------------------
// Kernel 4: out[src] += (ex/denom[src]) * V[dst]; 8 lanes per edge, float4
// gather of V, 4 fp32 L2 near-atomics per lane.
// ---------------------------------------------------------------------------
__global__ __launch_bounds__(256) void gat_aggregate(
    const float* __restrict__ V,
    const int* __restrict__ src, const int* __restrict__ dst,
    const float* __restrict__ ex, const float* __restrict__ denom,
    float* __restrict__ out, int E)
{
    const int e = blockIdx.x * 32 + (threadIdx.x >> 3);
    if (e >= E) return;
    const int sub = threadIdx.x & 7;
    const int s = src[e];
    const int d = dst[e];
    const float alpha = ex[e] / denom[s];

    const float4 v = ((const float4*)(V + (size_t)d * HEAD_SIZE))[sub];
    float* o = out + (size_t)s * HEAD_SIZE + sub * 4;
    atomicAdd(o + 0, alpha * v.x);
    atomicAdd(o + 1, alpha * v.y);
    atomicAdd(o + 2, alpha * v.z);
    atomicAdd(o + 3, alpha * v.w);
}

// ---------------------------------------------------------------------------
// Launch
// ---------------------------------------------------------------------------
extern "C" void kernel_launch(void* const* d_in, const int* in_sizes, int n_in,
                              void* d_out, int out_size, void* d_ws, size_t ws_size,
                              hipStream_t stream) {
    const float* X    = (const float*)d_in[0];
    const int*   edge = (const int*)  d_in[1];
    const float* Wq   = (const float*)d_in[2];
    const float* Wk   = (const float*)d_in[3];
    const float* Wv   = (const float*)d_in[4];

    const int N = in_sizes[0] / INPUT_SIZE;   // 100000
    const int E = in_sizes[1] / 2;            // 1600000
    const int* src = edge;                    // edge_index[0]
    const int* dst = edge + E;                // edge_index[1]

    float* out = (float*)d_out;

    // Workspace layout (floats): Q | K | V | scores/ex | denom | rmax(uint)
    float*    ws     = (float*)d_ws;
    float*    Q      = ws;
    float*    Kb     = ws + (size_t)N * HEAD_SIZE;
    float*    Vb     = ws + (size_t)2 * N * HEAD_SIZE;
    float*    scores = ws + (size_t)3 * N * HEAD_SIZE;
    float*    denom  = scores + E;
    unsigned* rmax   = (unsigned*)(denom + N);

    // 0: init out / denom / rowmax
    {
        const int M = (out_size > N) ? out_size : N;
        gat_init<<<(M + 255) / 256, 256, 0, stream>>>(out, denom, rmax, out_size, N);
    }
    // 1: QKV projection (fp32 WMMA); each wave does MTILES 16-row tiles,
    //    8 waves per block
    {
        const int tiles  = (N + 15) / 16;
        const int waves  = (tiles + MTILES - 1) / MTILES;
        const int blocks = (waves + 7) / 8;
        gat_qkv_wmma<<<blocks, 256, 0, stream>>>(X, Wq, Wk, Wv, Q, Kb, Vb, N);
    }
    // 2: edge scores + segment max
    gat_scores<<<(E + 31) / 32, 256, 0, stream>>>(Q, Kb, src, dst, scores, rmax, E);
    // 3: exp + segment sum
    gat_exp<<<(E + 255) / 256, 256, 0, stream>>>(src, scores, rmax, denom, E);
    // 4: weighted aggregation of V
    gat_aggregate<<<(E + 31) / 32, 256, 0, stream>>>(Vb, src, dst, scores, denom, out, E);
}